// QCONV1d_35880156791515
// MI455X (gfx1250) — compile-verified
//
#include <hip/hip_runtime.h>
#include <hip/hip_bf16.h>
#include <cmath>

// ---------------------------------------------------------------------------
// QCONV1d on MI455X (gfx1250).
// The 3-qubit VQC reduces analytically to q[b,w,k] =
//   cos(y[b,w+k,1]+vqc_w[b,w,0,1]) * cos(y[b,w+k,2]+vqc_w[b,w,0,2])
// so the pipeline is: f32 GEMM (x @ W_in^T), pointwise cos, rank-3 GEMM
// (q @ W_out^T + b_out).  Memory-bound: ~512MB HBM traffic -> ~22us @23.3TB/s.
// V_WMMA_F32_16X16X4_F32 keeps exact f32 precision on both GEMMs; x staging
// uses GLOBAL_LOAD_ASYNC_TO_LDS_B128 (ASYNCcnt) when the toolchain exposes it.
// ---------------------------------------------------------------------------

typedef float v2f __attribute__((ext_vector_type(2)));
typedef float v8f __attribute__((ext_vector_type(8)));

#define BQ      16
#define LQ      8192
#define CIN     512
#define COUT    512
#define NWINQ   8189          // ceil((L-K)/stride)
#define ROWS    (BQ * LQ)     // 131072 rows of x / y
#define KC      128           // K-chunk staged in LDS
#define TPAD    132           // KC + 4 dwords pad -> conflict-free ds_load_b64
#define WPAD    516           // 512 + 4 pad for W_in rows in LDS

#if __has_builtin(__builtin_amdgcn_global_load_async_to_lds_b128)
#define HAS_ASYNC_LDS 1
// Builtin signature (from hipcc diagnostic): arg0 = v4i in global AS,
// arg1 = LDS side, plus imm offset + imm cpol.
typedef int v4i_b __attribute__((vector_size(16)));
typedef __attribute__((address_space(1))) v4i_b* gv4p;
typedef __attribute__((address_space(3))) v4i_b* lv4p;
#else
#define HAS_ASYNC_LDS 0
#endif

// ---------------------------------------------------------------------------
// Kernel A: y[:, {1,2}] = (x @ W_in^T + b_in)[:, {1,2}]
// One wave owns a 16-row tile; 4 chunks of K=128 staged in LDS; 128 WMMAs.
// ---------------------------------------------------------------------------
__global__ __launch_bounds__(128) void qconv_yproj_wmma(
    const float* __restrict__ x, const float* __restrict__ W_in,
    const float* __restrict__ b_in, float* __restrict__ y1,
    float* __restrict__ y2)
{
    __shared__ float wbuf[4 * WPAD];          // W_in rows 0..2 + zero row 3
    __shared__ float tile[4][16 * TPAD];      // per-wave 16 x KC f32 tile

    const int tid  = threadIdx.x;
    const int wave = tid >> 5;
    const int lane = tid & 31;

    // Stage W_in^T source (padded with a zero row for columns n >= 3).
    for (int i = tid; i < 4 * WPAD; i += 128) {
        const int r = i / WPAD, c = i % WPAD;
        wbuf[i] = (r < 3 && c < CIN) ? W_in[r * CIN + c] : 0.0f;
    }
    __syncthreads();

    const int m0   = (blockIdx.x * 4 + wave) * 16;   // first row of this tile
    const int n    = lane & 15;                      // A row / B-D column
    const int half = lane >> 4;                      // 0: K=0,1   1: K=2,3
    const int kk   = half * 2;

    // C init = b_in broadcast down each column (zero for pad columns).
    const float cinit = (n < 3) ? b_in[n] : 0.0f;
    v8f acc;
#pragma unroll
    for (int r = 0; r < 8; ++r) acc[r] = cinit;

    float* myTile = tile[wave];
    for (int k0c = 0; k0c < CIN; k0c += KC) {
#if HAS_ASYNC_LDS
        // Async engine writes LDS without VGPR staging.  Async ops are NOT
        // ordered with DS ops: drain our ds_loads of the previous chunk
        // before the tile is overwritten, then drain the async copies before
        // the WMMA loop reads the tile.
        asm volatile("s_wait_dscnt 0" ::: "memory");
#pragma unroll
        for (int i = 0; i < 16; ++i) {
            __builtin_amdgcn_global_load_async_to_lds_b128(
                (gv4p)&x[(size_t)(m0 + i) * CIN + k0c + lane * 4],
                (lv4p)&myTile[i * TPAD + lane * 4], 0, 0);
        }
#if __has_builtin(__builtin_amdgcn_s_wait_asynccnt)
        __builtin_amdgcn_s_wait_asynccnt(0);
#else
        asm volatile("s_wait_asynccnt 0" ::: "memory");
#endif
#else
        // Fallback: coalesced b128 loads staged through VGPRs.
#pragma unroll
        for (int i = 0; i < 16; ++i) {
            const float4 v =
                *(const float4*)&x[(size_t)(m0 + i) * CIN + k0c + lane * 4];
            *(float4*)&myTile[i * TPAD + lane * 4] = v;
        }
#endif
        // 32 WMMAs over this chunk.  A: lane holds (A[m][kk], A[m][kk+1]);
        // B: lane holds (W_in[n][k+kk], W_in[n][k+kk+1]) (zero row for n>=3).
#pragma unroll
        for (int k = 0; k < KC; k += 4) {
            v2f a = *(const v2f*)&myTile[n * TPAD + k + kk];
            v2f b = *(const v2f*)&wbuf[min(n, 3) * WPAD + k0c + k + kk];
            acc = __builtin_amdgcn_wmma_f32_16x16x4_f32(
                false, a, false, b, (short)0, acc, false, false);
        }
    }

    // D layout: lane holds column N = lane&15, rows M = r + 8*half.
    if (n == 1 || n == 2) {
        float* dst = (n == 1) ? y1 : y2;
#pragma unroll
        for (int r = 0; r < 8; ++r)
            dst[m0 + r + 8 * half] = acc[r];
    }
}

// ---------------------------------------------------------------------------
// Kernel B: out[b,w,:] = q[b,w,:] @ W_out^T + b_out,
//           q[b,w,k] = cos(y1[w+k]+th1)*cos(y2[w+k]+th2)  (K padded 3->4)
// One wave owns a 16-window tile; 32 WMMAs cover the 512 output channels.
// Each half-wave store covers one aligned 64B sector -> full HBM efficiency.
// ---------------------------------------------------------------------------
__global__ __launch_bounds__(256) void qconv_out_wmma(
    const float* __restrict__ y1, const float* __restrict__ y2,
    const float* __restrict__ vqc_w, const float* __restrict__ W_out,
    const float* __restrict__ b_out, float* __restrict__ out)
{
    const int tid    = threadIdx.x;
    const int wave   = tid >> 5;
    const int lane   = tid & 31;
    const int tileId = blockIdx.x * 8 + wave;   // 0 .. 8191
    const int bb     = tileId >> 9;             // batch  (512 tiles per batch)
    const int w0     = (tileId & 511) * 16;     // first window of tile
    const int m      = lane & 15;               // A row (window) / D column
    const int half   = lane >> 4;
    const int kk     = half * 2;                // K pair this lane holds

    // Build A operand (q tile) in registers.
    const int   w   = w0 + m;
    const int   wc  = min(w, NWINQ - 1);        // clamp tail windows (masked)
    const float th1 = vqc_w[(size_t)(bb * NWINQ + wc) * 3 + 1];
    const float th2 = vqc_w[(size_t)(bb * NWINQ + wc) * 3 + 2];
    const int   base = bb * LQ;
    const int   r0  = base + min(w + kk, LQ - 1);
    const int   r1  = base + min(w + kk + 1, LQ - 1);

    v2f a;
    a.x = cosf(y1[r0] + th1) * cosf(y2[r0] + th2);
    a.y = (kk + 1 < 3) ? cosf(y1[r1] + th1) * cosf(y2[r1] + th2) : 0.0f;

#pragma unroll 2
    for (int n0 = 0; n0 < COUT; n0 += 16) {
        const int c = n0 + m;                   // output channel for this lane
        v2f b;
        b.x = W_out[c * 3 + kk];                // kk in {0,2} < 3 always
        b.y = (kk == 0) ? W_out[c * 3 + 1] : 0.0f;   // K=3 pad row = 0

        const float bo = b_out[c];
        v8f cc;
#pragma unroll
        for (int r = 0; r < 8; ++r) cc[r] = bo;

        cc = __builtin_amdgcn_wmma_f32_16x16x4_f32(
            false, a, false, b, (short)0, cc, false, false);

#pragma unroll
        for (int r = 0; r < 8; ++r) {
            const int ww = w0 + r + 8 * half;   // D row = window
            if (ww < NWINQ)
                out[((size_t)bb * NWINQ + ww) * COUT + c] = cc[r];
        }
    }
}

extern "C" void kernel_launch(void* const* d_in, const int* in_sizes, int n_in,
                              void* d_out, int out_size, void* d_ws, size_t ws_size,
                              hipStream_t stream) {
    const float* x     = (const float*)d_in[0];
    const float* W_in  = (const float*)d_in[1];
    const float* b_in  = (const float*)d_in[2];
    const float* W_out = (const float*)d_in[3];
    const float* b_out = (const float*)d_in[4];
    const float* vqc_w = (const float*)d_in[5];
    float* outp = (float*)d_out;

    float* y1 = (float*)d_ws;          // ROWS floats
    float* y2 = y1 + ROWS;             // ROWS floats (total 1 MB of d_ws)

    // Kernel A: 131072 rows / (16 rows * 4 waves) = 2048 blocks.
    qconv_yproj_wmma<<<ROWS / (16 * 4), 128, 0, stream>>>(x, W_in, b_in, y1, y2);

    // Kernel B: 16 batches * 512 window-tiles / 8 waves = 1024 blocks.
    qconv_out_wmma<<<(BQ * 512) / 8, 256, 0, stream>>>(y1, y2, vqc_w, W_out,
                                                       b_out, outp);
}